// MessagePassing_55439437856867
// MI455X (gfx1250) — compile-verified
//
#include <hip/hip_runtime.h>

#define N_NODES 4096
#define N_LAYERS 8
#define D 128

typedef float v2f __attribute__((ext_vector_type(2)));
typedef float v8f __attribute__((ext_vector_type(8)));

// ---------------------------------------------------------------------------
// src[l,j] = max i with (adj[l,i,j] != 0 or i == j).
// Descending scan with early exit: first hit from the top is the max.
// 4 rows per step so 4 independent loads (shared base + literal offsets) are
// in flight per dependent step.  No atomics: one thread owns each column.
// grid = (N/256 column chunks, 1, L); 256 threads, thread -> column j.
// ---------------------------------------------------------------------------
__global__ __launch_bounds__(256) void src_scan_kernel(const int* __restrict__ adj,
                                                       int* __restrict__ src) {
  const int l = blockIdx.z;
  const int j = blockIdx.x * 256 + threadIdx.x;
  const int* p = adj + ((size_t)l << 24) + j;   // column j of layer l
  int m = -1;
  for (int i = N_NODES - 4; i >= 0; i -= 4) {
    const int a3 = p[(size_t)(i + 3) << 12];
    const int a2 = p[(size_t)(i + 2) << 12];
    const int a1 = p[(size_t)(i + 1) << 12];
    const int a0 = p[(size_t)(i + 0) << 12];
    if (a3 | a2 | a1 | a0) {
      m = a3 ? (i + 3) : (a2 ? (i + 2) : (a1 ? (i + 1) : i));
      break;
    }
  }
  src[l * N_NODES + j] = (m > j) ? m : j;       // diagonal term
}

// ---------------------------------------------------------------------------
// deg[l*N + i] = (row sum of adj) + 1.  One wave per row: 32 lanes read
// int4s (512 B per coalesced b128 load), accumulate two columns per
// v_add_nc_u64 (values are 0/1, no carry across halves for <2^32 counts),
// then a 5-step shuffle reduction.  ~2 VALU per 16 bytes.
// grid = 2048 blocks x 256 threads; 8 waves/block, 2 rows/wave.
// ---------------------------------------------------------------------------
__global__ __launch_bounds__(256) void deg_kernel(const int* __restrict__ adj,
                                                  float* __restrict__ deg) {
  const int wave = threadIdx.x >> 5;
  const int lane = threadIdx.x & 31;
#pragma unroll
  for (int rr = 0; rr < 2; ++rr) {
    const int li = (blockIdx.x * 8 + wave) * 2 + rr;          // l*N + i
    const int4* row = (const int4*)(adj + ((size_t)li << 12)) + lane;
    unsigned long long acc = 0;
#pragma unroll 8
    for (int k = 0; k < 32; ++k) {
      const int4 v = row[k * 32];
      acc += ((unsigned long long)(unsigned)v.x) |
             (((unsigned long long)(unsigned)v.y) << 32);
      acc += ((unsigned long long)(unsigned)v.z) |
             (((unsigned long long)(unsigned)v.w) << 32);
    }
    unsigned sum = (unsigned)acc + (unsigned)(acc >> 32);
    for (int off = 16; off > 0; off >>= 1) sum += __shfl_xor(sum, off, 32);
    if (lane == 0) deg[li] = (float)sum + 1.0f;
  }
}

// ---------------------------------------------------------------------------
// Gather + GEMM + scale using V_WMMA_F32_16X16X4_F32 (wave32).
//
// Flattened output column t = j*L + l; out[t*128 + d] (contiguous per t).
// Each wave owns 16 columns (full D=128 output), block = 8 waves = 128 cols.
// W (128x128 f32, 64 KB) staged in LDS with an XOR swizzle at float2
// granularity so A-tile LDS loads (lane = M row) are bank-conflict free.
//
// WMMA operand layout (ISA 7.12.2, 16x16x4 f32):
//   A (16x4): lane m -> VGPR0 = A[m][k0 + 2*half], VGPR1 = A[m][k0+2*half+1]
//   B (4x16): lane n -> VGPR0 = B[k0 + 2*half][n], VGPR1 = B[k0+2*half+1][n]
//   C/D:      VGPR r -> (M = r + 8*half, N = lane&15)
// ---------------------------------------------------------------------------
extern __shared__ float w_lds[];                   // 128*128 floats, swizzled

__global__ __launch_bounds__(256) void gnn_wmma_kernel(
    const float* __restrict__ feature, const float* __restrict__ Wm,
    const int* __restrict__ src, const float* __restrict__ deg,
    float* __restrict__ out) {
  const int tid = threadIdx.x;

  // Stage W into LDS, XOR-swizzled on float2 granularity.
  for (int idx = tid; idx < D * D; idx += 256) {
    const int row = idx >> 7;
    const int kk  = idx & 127;
    const int swz = ((((kk >> 1) ^ ((row & 15) << 1)) << 1) | (kk & 1));
    w_lds[(row << 7) + swz] = Wm[idx];
  }
  __syncthreads();

  const int wave = tid >> 5;
  const int lane = tid & 31;
  const int n    = lane & 15;                      // column within wave tile
  const int half = lane >> 4;                      // K-pair selector

  const int t = (blockIdx.x * 8 + wave) * 16 + n;  // flattened (j,l)
  const int j = t >> 3;                            // t / L
  const int l = t & 7;                             // t % L
  const int s = src[l * N_NODES + j];
  const float sc = rsqrtf(deg[l * N_NODES + j] * deg[l * N_NODES + s]);
  const float* fptr = feature + (((size_t)s * N_LAYERS + l) << 7) + (half << 1);

  v8f acc[8] = {};                                 // 8 M-tiles x 8 VGPRs

  for (int k = 0; k < D; k += 4) {
    v2f b;                                         // gathered feature fragment
    b.x = fptr[k];
    b.y = fptr[k + 1];
    const int kpair = (k >> 1) + half;
#pragma unroll
    for (int m = 0; m < 8; ++m) {
      const int row = (m << 4) + n;
      const v2f a = *(const v2f*)&w_lds[(row << 7) + ((kpair ^ (n << 1)) << 1)];
      acc[m] = __builtin_amdgcn_wmma_f32_16x16x4_f32(
          /*neg_a=*/false, a, /*neg_b=*/false, b,
          /*c_mod=*/(short)0, acc[m], /*reuse_a=*/false, /*reuse_b=*/false);
    }
  }

  float* outp = out + ((size_t)t << 7) + (half << 3);
#pragma unroll
  for (int m = 0; m < 8; ++m) {
#pragma unroll
    for (int r = 0; r < 8; ++r) {
      outp[(m << 4) + r] = acc[m][r] * sc;
    }
  }
}

// ---------------------------------------------------------------------------
extern "C" void kernel_launch(void* const* d_in, const int* in_sizes, int n_in,
                              void* d_out, int out_size, void* d_ws, size_t ws_size,
                              hipStream_t stream) {
  const float* feature = (const float*)d_in[0];    // [N, L, D] f32
  const float* Wm      = (const float*)d_in[1];    // [D, D]    f32
  const int*   adj     = (const int*)d_in[2];      // [L, N, N] i32
  float* out = (float*)d_out;                      // [N, L, D] f32

  int*   src = (int*)d_ws;                                               // L*N ints
  float* deg = (float*)((char*)d_ws + sizeof(int) * N_LAYERS * N_NODES); // L*N floats

  src_scan_kernel<<<dim3(N_NODES / 256, 1, N_LAYERS), 256, 0, stream>>>(adj, src);
  deg_kernel<<<(N_LAYERS * N_NODES) / 16, 256, 0, stream>>>(adj, deg);
  gnn_wmma_kernel<<<(N_NODES * N_LAYERS) / 128, 256, D * D * sizeof(float),
                    stream>>>(feature, Wm, src, deg, out);
}